// PrePrompt_34359738990
// MI455X (gfx1250) — compile-verified
//
#include <hip/hip_runtime.h>
#include <math.h>
#include <stdint.h>

// CDNA5 / gfx1250: wave32, f32 WMMA Gram-matrix core + async global->LDS
// double-buffered gather pipeline (ASYNCcnt), no VGPR staging.

typedef __attribute__((ext_vector_type(2))) float v2f;
typedef __attribute__((ext_vector_type(8))) float v8f;

#define NBLOCKS     4096
#define DIM         256
#define TUP         10
#define ROWS_PITCH  260   // 260 % 64 = 4 -> conflict-free strided lane reads

// Each lane moves 16B; two B128 async ops cover one 1KB row.
// IOFFSET is added to BOTH the LDS dest and the global source address.
__device__ __forceinline__ void async_row_to_lds(const float* gsrc,
                                                 const float* lds_dst,
                                                 int lane)
{
    uint32_t l = (uint32_t)(uintptr_t)lds_dst + (uint32_t)(lane * 16);
    uint64_t g = (uint64_t)(uintptr_t)gsrc + (uint64_t)(lane * 16);
    asm volatile("global_load_async_to_lds_b128 %0, %1, off"
                 :: "v"(l), "v"(g) : "memory");
    asm volatile("global_load_async_to_lds_b128 %0, %1, off offset:512"
                 :: "v"(l), "v"(g) : "memory");
}

// Issue the 11-row panel (10 gathered tuple rows + anchor row) for anchor i.
__device__ __forceinline__ void issue_panel(const float* __restrict__ feat,
                                            const int*   __restrict__ tup,
                                            float (*rowbuf)[ROWS_PITCH],
                                            int i, int lane)
{
    #pragma unroll
    for (int r = 0; r <= TUP; ++r) {
        int idx = (r == TUP) ? i : tup[(size_t)i * TUP + r];
        async_row_to_lds(feat + (size_t)idx * DIM, rowbuf[r], lane);
    }
}

// One wave32 per anchor i (grid-stride). Computes the 16x16 f32 Gram matrix of
// S = [t0..t9, h_i, 0^5] via 64 chained V_WMMA_F32_16X16X4_F32: G[j][j]=|t_j|^2,
// G[j][10]=dot(t_j,h_i), G[10][10]=|h_i|^2 -> all softmax-log inputs in one pass.
__global__ __launch_bounds__(32) void tuple_infonce_kernel(
    const float* __restrict__ feat,
    const int*   __restrict__ tup,
    float*       __restrict__ partials,
    int M)
{
    __shared__ __align__(16) float rows[2][16][ROWS_PITCH];
    __shared__ __align__(16) float gram[16 * 16];

    const int lane = threadIdx.x;   // 0..31
    const int half = lane >> 4;     // 0 or 1
    const int r16  = lane & 15;

    // Rows 11..15 of both panels are zero padding; written once.
    for (int b = 0; b < 2; ++b)
        for (int r = 11; r < 16; ++r)
            for (int c = lane; c < DIM; c += 32)
                rows[b][r][c] = 0.0f;
    __syncthreads();

    float partial = 0.0f;

    if ((int)blockIdx.x < M) {
        int buf = 0;
        issue_panel(feat, tup, rows[0], blockIdx.x, lane);   // prologue fill

        for (int i = blockIdx.x; i < M; i += NBLOCKS) {
            // Overlap: start filling the other panel for the next anchor,
            // then wait only for the CURRENT panel (async ops retire in order).
            int inext = i + NBLOCKS;
            if (inext < M) {
                issue_panel(feat, tup, rows[buf ^ 1], inext, lane);
                asm volatile("s_wait_asynccnt 0x16" ::: "memory"); // <=22 left
            } else {
                asm volatile("s_wait_asynccnt 0x0" ::: "memory");  // drain
            }
            __syncthreads();

            // ---- G = S*S^T, K=256 in 64 x V_WMMA_F32_16X16X4_F32 ----
            // A fragment: lane l, vgpr v -> S[l&15][4k + 2*(l>>4) + v]; the
            // B fragment of S^T maps to identical per-lane values (symmetry),
            // so one ds_load_2addr_b64 feeds both operands.
            const float* prow = rows[buf][r16];
            v8f acc = {0.f, 0.f, 0.f, 0.f, 0.f, 0.f, 0.f, 0.f};
            for (int k = 0; k < DIM / 4; ++k) {
                int kk = 4 * k + 2 * half;
                v2f a;
                a.x = prow[kk];
                a.y = prow[kk + 1];
                acc = __builtin_amdgcn_wmma_f32_16x16x4_f32(
                    false, a, false, a, (short)0, acc, false, false);
            }

            // ---- spill D: lane l, vgpr v holds G[v + 8*(l>>4)][l&15] ----
            #pragma unroll
            for (int v = 0; v < 8; ++v)
                gram[(v + 8 * half) * 16 + r16] = acc[v];
            __syncthreads();

            // ---- per-lane j (only j=0..9 meaningful), clamped reads ----
            int j = r16;
            float gjj = gram[j * 16 + j];        // ||t_j||^2
            float gji = gram[j * 16 + TUP];      // dot(t_j, h_i)
            float nii = gram[TUP * 16 + TUP];    // ||h_i||^2
            float nrm = fmaxf(sqrtf(gjj), 1e-8f) * fmaxf(sqrtf(nii), 1e-8f);
            float sim = gji / nrm;
            float e   = __expf(sim);

            // denominator = sum of e over LANES 1..9, deterministic xor tree
            float ed = (lane >= 1 && lane <= 9) ? e : 0.0f;
            #pragma unroll
            for (int m = 16; m >= 1; m >>= 1)
                ed += __shfl_xor(ed, m, 32);
            float sim0 = __shfl(sim, 0, 32);

            partial += __logf(ed) - sim0;        // -log(e0 / den)
            __syncthreads();                     // gram reused next iter
            buf ^= 1;
        }
    }

    if (lane == 0) partials[blockIdx.x] = partial;
}

// Deterministic final reduction: fixed strides, fixed tree order.
__global__ __launch_bounds__(256) void reduce_kernel(
    const float* __restrict__ partials, int nb, float* __restrict__ out, float inv_m)
{
    __shared__ float s[256];
    float acc = 0.0f;
    for (int j = threadIdx.x; j < nb; j += 256)
        acc += partials[j];
    s[threadIdx.x] = acc;
    __syncthreads();
    for (int off = 128; off > 0; off >>= 1) {
        if ((int)threadIdx.x < off) s[threadIdx.x] += s[threadIdx.x + off];
        __syncthreads();
    }
    if (threadIdx.x == 0) out[0] = s[0] * inv_m;
}

extern "C" void kernel_launch(void* const* d_in, const int* in_sizes, int n_in,
                              void* d_out, int out_size, void* d_ws, size_t ws_size,
                              hipStream_t stream) {
    const float* feat = (const float*)d_in[0];   // (200000, 256) f32
    const int*   tup  = (const int*)d_in[1];     // (50000, 10) int
    int M = in_sizes[1] / TUP;

    float* partials = (float*)d_ws;              // NBLOCKS floats of scratch

    tuple_infonce_kernel<<<NBLOCKS, 32, 0, stream>>>(feat, tup, partials, M);
    reduce_kernel<<<1, 256, 0, stream>>>(partials, NBLOCKS, (float*)d_out,
                                         1.0f / (float)M);
}